// DenseCapsule_3186865734189
// MI455X (gfx1250) — compile-verified
//
#include <hip/hip_runtime.h>

// ---------------------------------------------------------------------------
// DenseCapsule dynamic routing, MI455X (gfx1250, wave32, WMMA).
//
// Shapes: B=64, IN_N=1152, IN_D=16, OUT_N=64, OUT_D=32, ROUTINGS=3.
// Pipeline:
//   0) memset S0,S1,S2 accumulators (one hipMemsetAsync, capture-safe)
//   1) k_gemm : x_hat = W @ x  via V_WMMA_F32_16X16X4_F32,
//               stores x_hat as f16 [b][i][o][d] in ws (302 MB),
//               accumulates S0[b,o,d] = sum_i x_hat (fp32, atomics)
//   2) k_squash(S0 * 1/64) -> outputs0      (round 0: softmax(0) is uniform)
//   3) k_route(outputs0, b=0)  -> writes logits T=b1, accumulates S1
//   4) k_squash(S1) -> outputs1
//   5) k_route(outputs1, b=T)  -> accumulates S2 (no logit write needed)
//   6) k_squash(S2) -> d_out
// ---------------------------------------------------------------------------

#define BATCH  64
#define IN_N   1152
#define IN_D   16
#define OUT_N  64
#define OUT_D  32

#define GSPLIT 18                 // i-slabs per o in the GEMM
#define GIPB   (IN_N / GSPLIT)    // 64 i's per workgroup
#define RCH    8                  // i-chunks per b in routing
#define RIPB   (IN_N / RCH)       // 144 i's per routing block (8 waves * 18)

typedef float     v2f __attribute__((ext_vector_type(2)));
typedef float     v8f __attribute__((ext_vector_type(8)));
typedef _Float16  h8  __attribute__((ext_vector_type(8)));

// ---------------------------------------------------------------------------
// Kernel 1: x_hat GEMM with V_WMMA_F32_16X16X4_F32.
// Per workgroup: one o, one slab of GIPB i's, all 64 b, all 32 d.
// 8 waves = 2 M-tiles (d: 0..15, 16..31) x 4 N-tiles (b: 16 each).
// Per i: 4 K-steps of 16x16x4 fp32 WMMA. Epilogue: packed f16 b128 store of
// x_hat + running fp32 accumulation of S0 = sum_i x_hat.
// ---------------------------------------------------------------------------
__global__ __launch_bounds__(256) void k_gemm(const float* __restrict__ W,
                                              const float* __restrict__ X,
                                              _Float16* __restrict__ XH,
                                              float* __restrict__ S0) {
  const int o    = blockIdx.x;
  const int i0   = blockIdx.y * GIPB;
  const int lane = threadIdx.x & 31;
  const int wave = threadIdx.x >> 5;
  const int d0   = (wave >> 2) * 16;   // M-tile origin (d)
  const int b0   = (wave &  3) * 16;   // N-tile origin (b)
  const int hl   = lane >> 4;          // half-wave: selects K=2*hl+{0,1}
  const int l16  = lane & 15;

  const float* Wo = W + (size_t)o * IN_N * OUT_D * IN_D;
  v8f sacc = {};                       // per-lane S0 partial (8 rows)

  for (int ii = 0; ii < GIPB; ++ii) {
    const int i = i0 + ii;
    // A fragment source: W[o][i][d0+l16][2*hl + 4*s + {0,1}]  (ISA 16x4 f32 layout)
    const float* wa = Wo + (size_t)i * OUT_D * IN_D + (d0 + l16) * IN_D + 2 * hl;
    // B fragment source: x[b0+l16][i][2*hl + 4*s + {0,1}]     (4x16 f32 layout)
    const float* xb = X + ((size_t)(b0 + l16) * IN_N + i) * IN_D + 2 * hl;

    __builtin_prefetch(wa + OUT_D * IN_D, 0, 1);   // global_prefetch next i's W row

    v8f c = {};
#pragma unroll
    for (int s = 0; s < 4; ++s) {
      v2f a  = *(const v2f*)(wa + 4 * s);
      v2f bf = *(const v2f*)(xb + 4 * s);
      // D = A(16x4) * B(4x16) + C ; fp32 in / fp32 out
      c = __builtin_amdgcn_wmma_f32_16x16x4_f32(false, a, false, bf,
                                                (short)0, c, false, false);
    }
    sacc += c;

    // C layout: vgpr v holds rows (d0+v) for lanes 0-15 and (d0+v+8) for 16-31,
    // column b = b0 + l16. Lane's 8 rows are consecutive d -> pack one b128.
    h8 hv;
#pragma unroll
    for (int v = 0; v < 8; ++v) hv[v] = (_Float16)c[v];
    const int b      = b0 + l16;
    const int dstart = d0 + 8 * hl;
    *(h8*)(XH + (((size_t)b * IN_N + i) * OUT_N + o) * OUT_D + dstart) = hv;
  }

  // S0[b][o][d] += sum over this i-slab
  const int b      = b0 + l16;
  const int dstart = d0 + 8 * hl;
  float* sp = S0 + ((size_t)b * OUT_N + o) * OUT_D + dstart;
#pragma unroll
  for (int v = 0; v < 8; ++v) atomicAdd(sp + v, sacc[v]);
}

// ---------------------------------------------------------------------------
// Kernel 2: fused routing pass. One wave owns one (b, i): lane L handles
// o = L and o = L+32. Computes t = b_prev + outputs_prev . x_hat, softmax over
// the 64 o's with wave32 __shfl_xor allreduce (no barriers), optionally writes
// the new logits, and accumulates c * x_hat into registers -> LDS -> global.
// ---------------------------------------------------------------------------
__global__ __launch_bounds__(256) void k_route(const _Float16* __restrict__ XH,
                                               const float* __restrict__ OPREV,
                                               const float* __restrict__ BPREV,  // may be null (b=0)
                                               float* __restrict__ BNEXT,        // may be null
                                               float* __restrict__ SOUT) {
  __shared__ float op_lds[OUT_N * OUT_D];  // outputs_prev[b]  (8 KB)
  __shared__ float s_lds[OUT_N * OUT_D];   // block S partial  (8 KB)

  const int b  = blockIdx.x;
  const int ch = blockIdx.y;
  const int t  = threadIdx.x;

  for (int j = t; j < OUT_N * OUT_D; j += 256) {
    op_lds[j] = OPREV[(size_t)b * OUT_N * OUT_D + j];
    s_lds[j]  = 0.f;
  }
  __syncthreads();

  const int lane = t & 31, wave = t >> 5;
  const int o0 = lane, o1 = lane + 32;

  float acc0[OUT_D], acc1[OUT_D];
#pragma unroll
  for (int d = 0; d < OUT_D; ++d) { acc0[d] = 0.f; acc1[d] = 0.f; }

  for (int j = 0; j < RIPB / 8; ++j) {
    const int i = ch * RIPB + j * 8 + wave;
    const size_t base = ((size_t)b * IN_N + i) * OUT_N;
    const h8* p0 = (const h8*)(XH + (base + o0) * OUT_D);
    const h8* p1 = (const h8*)(XH + (base + o1) * OUT_D);

    float x0[OUT_D], x1[OUT_D];
#pragma unroll
    for (int q = 0; q < 4; ++q) {
      h8 h0 = p0[q], h1 = p1[q];
#pragma unroll
      for (int e = 0; e < 8; ++e) {
        x0[q * 8 + e] = (float)h0[e];
        x1[q * 8 + e] = (float)h1[e];
      }
    }

    float t0 = 0.f, t1 = 0.f;
#pragma unroll
    for (int d = 0; d < OUT_D; ++d) {
      t0 += op_lds[o0 * OUT_D + d] * x0[d];
      t1 += op_lds[o1 * OUT_D + d] * x1[d];
    }
    if (BPREV) {
      t0 += BPREV[base + o0];
      t1 += BPREV[base + o1];
    }
    if (BNEXT) {                // new logits b_next = b_prev + outputs.x_hat
      BNEXT[base + o0] = t0;
      BNEXT[base + o1] = t1;
    }

    // softmax over 64 o's: 2 per lane + wave32 xor-allreduce
    float m = fmaxf(t0, t1);
#pragma unroll
    for (int s = 16; s > 0; s >>= 1) m = fmaxf(m, __shfl_xor(m, s, 32));
    float e0 = __expf(t0 - m), e1 = __expf(t1 - m);
    float sum = e0 + e1;
#pragma unroll
    for (int s = 16; s > 0; s >>= 1) sum += __shfl_xor(sum, s, 32);
    const float inv = 1.f / sum;
    const float c0 = e0 * inv, c1 = e1 * inv;

#pragma unroll
    for (int d = 0; d < OUT_D; ++d) {
      acc0[d] += c0 * x0[d];
      acc1[d] += c1 * x1[d];
    }
  }

  // registers -> LDS (ds_add_f32) -> global (global_atomic_add_f32)
#pragma unroll
  for (int d = 0; d < OUT_D; ++d) {
    atomicAdd(&s_lds[o0 * OUT_D + d], acc0[d]);
    atomicAdd(&s_lds[o1 * OUT_D + d], acc1[d]);
  }
  __syncthreads();
  for (int j = t; j < OUT_N * OUT_D; j += 256)
    atomicAdd(&SOUT[(size_t)b * OUT_N * OUT_D + j], s_lds[j]);
}

// ---------------------------------------------------------------------------
// Kernel 3: squash. out[b,o,:] = squash(S[b,o,:] * prescale)
// ---------------------------------------------------------------------------
__global__ __launch_bounds__(64) void k_squash(const float* __restrict__ S,
                                               float* __restrict__ OUT,
                                               float prescale) {
  const int b = blockIdx.x, o = threadIdx.x;
  const float* sp = S + ((size_t)b * OUT_N + o) * OUT_D;
  float v[OUT_D];
  float s2 = 0.f;
#pragma unroll
  for (int d = 0; d < OUT_D; ++d) {
    v[d] = sp[d] * prescale;
    s2 += v[d] * v[d];
  }
  const float n    = sqrtf(s2);
  const float coef = s2 / (1.f + s2) / (n + 1e-8f);
  float* op = OUT + ((size_t)b * OUT_N + o) * OUT_D;
#pragma unroll
  for (int d = 0; d < OUT_D; ++d) op[d] = coef * v[d];
}

// ---------------------------------------------------------------------------
extern "C" void kernel_launch(void* const* d_in, const int* in_sizes, int n_in,
                              void* d_out, int out_size, void* d_ws, size_t ws_size,
                              hipStream_t stream) {
  const float* X = (const float*)d_in[0];   // [64,1152,16]
  const float* W = (const float*)d_in[1];   // [64,1152,32,16]
  float* OUT = (float*)d_out;               // [64,64,32]

  char* ws = (char*)d_ws;
  const size_t XH_BYTES = (size_t)BATCH * IN_N * OUT_N * OUT_D * sizeof(_Float16); // 302 MB
  const size_t T_BYTES  = (size_t)BATCH * IN_N * OUT_N * sizeof(float);            // 18.9 MB
  const size_t S_BYTES  = (size_t)BATCH * OUT_N * OUT_D * sizeof(float);           // 512 KB

  size_t off = 0;
  _Float16* XH = (_Float16*)(ws + off); off += XH_BYTES;
  float* T  = (float*)(ws + off); off += T_BYTES;
  float* S0 = (float*)(ws + off); off += S_BYTES;
  float* S1 = (float*)(ws + off); off += S_BYTES;
  float* S2 = (float*)(ws + off); off += S_BYTES;
  float* O0 = (float*)(ws + off); off += S_BYTES;
  float* O1 = (float*)(ws + off); off += S_BYTES;

  // Zero the three atomic accumulators (S0,S1,S2 are contiguous).
  hipMemsetAsync(S0, 0, 3 * S_BYTES, stream);

  // 1) x_hat GEMM (WMMA fp32) + S0 accumulation
  k_gemm<<<dim3(OUT_N, GSPLIT), 256, 0, stream>>>(W, X, XH, S0);
  // 2) round 0: c uniform -> outputs0 = squash(S0/64)
  k_squash<<<BATCH, OUT_N, 0, stream>>>(S0, O0, 1.f / 64.f);
  // 3) round 1 fused: b1 = outputs0.x_hat ; c1=softmax(b1) ; S1 = sum c1*x_hat
  k_route<<<dim3(BATCH, RCH), 256, 0, stream>>>(XH, O0, nullptr, T, S1);
  k_squash<<<BATCH, OUT_N, 0, stream>>>(S1, O1, 1.f);
  // 4) round 2 fused (final): b2 = b1 + outputs1.x_hat ; S2 = sum softmax(b2)*x_hat
  k_route<<<dim3(BATCH, RCH), 256, 0, stream>>>(XH, O1, T, nullptr, S2);
  k_squash<<<BATCH, OUT_N, 0, stream>>>(S2, OUT, 1.f);
}